// PyramidTextEncoder_40329742909526
// MI455X (gfx1250) — compile-verified
//
#include <hip/hip_runtime.h>
#include <math.h>

// ---------------------------------------------------------------------------
// PyramidTextEncoder forward for MI455X (gfx1250, wave32, WMMA + TDM).
//
// Big GEMMs: 8-wave blocks, 64x128 macro-tiles, K staged through LDS in
// 32-wide chunks by the Tensor Data Mover (tensor_load_to_lds), double
// buffered, synchronized with s_wait_tensorcnt + barriers.  TDM padding
// (2 dwords per 32) gives a conflict-free LDS stride of 34.
// Small GEMMs: wave-per-tile, register double-buffered so loads overlap WMMA.
// All matrix math is V_WMMA_F32_16X16X4_F32 (exact f32, matches reference).
// ---------------------------------------------------------------------------

#define BB     32
#define LL     512
#define DD     768
#define HID    192
#define KK     5
#define DI     1536
#define DS     16
#define DTR    48
#define DCONV  4
#define NROWS  (BB * LL)     // 16384
#define EPSV   1e-5f

#define BM   64
#define BN   128
#define KC   32
#define KCP  34   // KC + 2 pad dwords (TDM pad_interval=32dw, pad_amount=2dw)

typedef float v2f __attribute__((ext_vector_type(2)));
typedef float v8f __attribute__((ext_vector_type(8)));
typedef unsigned int uivec4 __attribute__((ext_vector_type(4)));
typedef int ivec4 __attribute__((ext_vector_type(4)));
typedef int ivec8 __attribute__((ext_vector_type(8)));

__device__ __forceinline__ float sigmoid_(float x) { return 1.0f / (1.0f + __expf(-x)); }
__device__ __forceinline__ float silu_(float x)    { return x * sigmoid_(x); }
__device__ __forceinline__ float softplus_(float x){ return (x > 20.0f) ? x : log1pf(__expf(x)); }

// ---------------------------------------------------------------------------
// TDM: async 2D tile load global->LDS. D# per CDNA5 ISA:
//  group0: count=1 | lds_addr | global_addr | type=2
//  group1: data_size=4B, pad_enable (2 dwords every 32), dims/strides (elems)
//  groups2/3: zero (2D tensor).
// ---------------------------------------------------------------------------
__device__ __forceinline__ void tdm_load_2d(unsigned lds_addr, const float* gptr,
                                            unsigned tile_d0, unsigned tile_d1,
                                            unsigned long long stride_elems) {
  unsigned long long ga = (unsigned long long)(uintptr_t)gptr;
  uivec4 g0;
  g0[0] = 1u;                                   // count=1, user descriptor
  g0[1] = lds_addr;                             // LDS byte address
  g0[2] = (unsigned)(ga & 0xffffffffull);       // global_addr[31:0]
  g0[3] = (unsigned)((ga >> 32) & 0x1ffffffull) | (2u << 30);  // addr[56:32]|type=2
  ivec8 g1;
  g1[0] = (int)((2u << 16) | (1u << 20) | (4u << 22) | (1u << 25));
  //            data_size=4B  pad_en      intv=32dw    amount=2dw
  g1[1] = (int)((tile_d0 & 0xffffu) << 16);                       // tensor_dim0 lo
  g1[2] = (int)((tile_d0 >> 16) | ((tile_d1 & 0xffffu) << 16));   // tdim0 hi|tdim1 lo
  g1[3] = (int)((tile_d1 >> 16) | (tile_d0 << 16));               // tdim1 hi|tile_dim0
  g1[4] = (int)(tile_d1 & 0xffffu);                               // tile_dim1 (tile_dim2=0)
  g1[5] = (int)(unsigned)(stride_elems & 0xffffffffull);          // dim0_stride lo32
  g1[6] = (int)(unsigned)((stride_elems >> 32) & 0xffffull);      // dim0_stride hi16
  g1[7] = 0;
  ivec4 z4 = {0, 0, 0, 0};
#if __clang_major__ >= 23
  ivec8 z8 = {0, 0, 0, 0, 0, 0, 0, 0};
  __builtin_amdgcn_tensor_load_to_lds(g0, g1, z4, z4, z8, 0);
#else
  __builtin_amdgcn_tensor_load_to_lds(g0, g1, z4, z4, 0);
#endif
}

__device__ __forceinline__ unsigned lds_off_(const void* p) {
  // generic pointers to LDS carry the LDS byte offset in the low 32 bits
  return (unsigned)(uintptr_t)p;
}

// ---------------------------------------------------------------------------
// LDS-tiled GEMM: C[M,N] = A[M,K] * W[N,K]^T, requires M%64==0, N%128==0,
// K%32==0.  block = 256 threads = 8 waves (4 M-subtiles x 2 N-halves);
// each wave computes 16x64 via 4 accumulators.  TDM double-buffers A/B.
// epi: 0 none, 1 softplus(v+bias[col]), 2 sigmoid(v+bias[col]).
// ---------------------------------------------------------------------------
__global__ void __launch_bounds__(256)
gemm_f32_wmma_lds(const float* __restrict__ A, const float* __restrict__ W,
                  float* __restrict__ C, int K, int lda, int ldw, int ldc,
                  const float* __restrict__ bias, int epi) {
  __shared__ float As[2][BM * KCP];
  __shared__ float Bs[2][BN * KCP];

  const int tid  = threadIdx.x;
  const int lane = tid & 31;
  const int wave = tid >> 5;
  const int r  = lane & 15;
  const int hi = lane >> 4;
  const int wm = wave & 3;     // M subtile 0..3
  const int wn = wave >> 2;    // N half    0..1
  const int m0 = blockIdx.y * BM;
  const int n0 = blockIdx.x * BN;

  const unsigned asAddr[2] = { lds_off_(&As[0][0]), lds_off_(&As[1][0]) };
  const unsigned bsAddr[2] = { lds_off_(&Bs[0][0]), lds_off_(&Bs[1][0]) };

  v8f acc[4];
#pragma unroll
  for (int j = 0; j < 4; ++j)
#pragma unroll
    for (int i = 0; i < 8; ++i) acc[j][i] = 0.0f;

  const int nk = K / KC;
  if (tid < 32) {  // wave 0 drives the TDM
    tdm_load_2d(asAddr[0], A + (size_t)m0 * lda, KC, BM, (unsigned long long)lda);
    tdm_load_2d(bsAddr[0], W + (size_t)n0 * ldw, KC, BN, (unsigned long long)ldw);
  }

  for (int kc = 0; kc < nk; ++kc) {
    const int buf = kc & 1;
    if (tid < 32) {
      if (kc + 1 < nk) {
        const size_t ko = (size_t)(kc + 1) * KC;
        tdm_load_2d(asAddr[buf ^ 1], A + (size_t)m0 * lda + ko, KC, BM,
                    (unsigned long long)lda);
        tdm_load_2d(bsAddr[buf ^ 1], W + (size_t)n0 * ldw + ko, KC, BN,
                    (unsigned long long)ldw);
        __builtin_amdgcn_s_wait_tensorcnt(2);  // chunk kc landed, kc+1 in flight
      } else {
        __builtin_amdgcn_s_wait_tensorcnt(0);
      }
    }
    __syncthreads();

    const float* abase = &As[buf][(wm * 16 + r) * KCP + 2 * hi];
    const float* bbase = &Bs[buf][(wn * 64 + r) * KCP + 2 * hi];
#pragma unroll
    for (int kk = 0; kk < KC; kk += 4) {
      v2f a = *(const v2f*)(abase + kk);
#pragma unroll
      for (int j = 0; j < 4; ++j) {
        v2f bf = *(const v2f*)(bbase + j * 16 * KCP + kk);
        acc[j] = __builtin_amdgcn_wmma_f32_16x16x4_f32(false, a, false, bf,
                                                       (short)0, acc[j], false, false);
      }
    }
    __syncthreads();
  }

#pragma unroll
  for (int j = 0; j < 4; ++j) {
    const int col = n0 + wn * 64 + j * 16 + r;
    const float bv = (epi != 0) ? bias[col] : 0.0f;
#pragma unroll
    for (int i = 0; i < 8; ++i) {
      const int row = m0 + wm * 16 + i + 8 * hi;
      float v = acc[j][i];
      if (epi == 1)      v = softplus_(v + bv);
      else if (epi == 2) v = sigmoid_(v + bv);
      C[(size_t)row * ldc + col] = v;
    }
  }
}

// ---------------------------------------------------------------------------
// Wave-per-tile GEMM, register double-buffered (loads for k+4 issued before
// the WMMAs for k).  For shapes the LDS kernel can't take.
// ---------------------------------------------------------------------------
template <int NT>
__global__ void gemm_f32_wmma_pipe(const float* __restrict__ A, const float* __restrict__ W,
                                   float* __restrict__ C, int K, int lda, int ldw, int ldc,
                                   const float* __restrict__ bias, int epi) {
  const int lane = threadIdx.x;
  const int r  = lane & 15;
  const int hi = lane >> 4;
  const int m0 = blockIdx.y * 16;
  const int n0 = blockIdx.x * (16 * NT);

  const float* aRow = A + (size_t)(m0 + r) * lda + 2 * hi;
  const float* wRow[NT];
#pragma unroll
  for (int j = 0; j < NT; ++j)
    wRow[j] = W + (size_t)(n0 + j * 16 + r) * ldw + 2 * hi;

  v8f acc[NT];
#pragma unroll
  for (int j = 0; j < NT; ++j)
#pragma unroll
    for (int i = 0; i < 8; ++i) acc[j][i] = 0.0f;

  v2f aCur = *(const v2f*)(aRow);
  v2f bCur[NT];
#pragma unroll
  for (int j = 0; j < NT; ++j) bCur[j] = *(const v2f*)(wRow[j]);

  for (int k0 = 0; k0 + 4 < K; k0 += 4) {
    if ((k0 & 63) == 0) __builtin_prefetch(aRow + k0 + 64, 0, 1);
    v2f aNxt = *(const v2f*)(aRow + k0 + 4);
    v2f bNxt[NT];
#pragma unroll
    for (int j = 0; j < NT; ++j) bNxt[j] = *(const v2f*)(wRow[j] + k0 + 4);
#pragma unroll
    for (int j = 0; j < NT; ++j)
      acc[j] = __builtin_amdgcn_wmma_f32_16x16x4_f32(false, aCur, false, bCur[j],
                                                     (short)0, acc[j], false, false);
    aCur = aNxt;
#pragma unroll
    for (int j = 0; j < NT; ++j) bCur[j] = bNxt[j];
  }
#pragma unroll
  for (int j = 0; j < NT; ++j)
    acc[j] = __builtin_amdgcn_wmma_f32_16x16x4_f32(false, aCur, false, bCur[j],
                                                   (short)0, acc[j], false, false);

#pragma unroll
  for (int j = 0; j < NT; ++j) {
    const int col = n0 + j * 16 + r;
    const float bv = (epi != 0) ? bias[col] : 0.0f;
#pragma unroll
    for (int i = 0; i < 8; ++i) {
      const int row = m0 + i + 8 * hi;
      float v = acc[j][i];
      if (epi == 1)      v = softplus_(v + bv);
      else if (epi == 2) v = sigmoid_(v + bv);
      C[(size_t)row * ldc + col] = v;
    }
  }
}

// ---------------------------------------------------------------------------
// BatchNorm statistics; emits fused scale = g*rstd, shift = b - mean*scale.
// ---------------------------------------------------------------------------
__global__ void bn_stats(const float* __restrict__ in, int C, int nrows,
                         const float* __restrict__ g, const float* __restrict__ b,
                         float* __restrict__ scale, float* __restrict__ shift) {
  const int c = blockIdx.x, tid = threadIdx.x;
  float s = 0.0f, q = 0.0f;
  for (int i = tid; i < nrows; i += 256) {
    float v = in[(size_t)i * C + c];
    s += v; q += v * v;
  }
  __shared__ float rs[256], rq[256];
  rs[tid] = s; rq[tid] = q; __syncthreads();
  for (int off = 128; off > 0; off >>= 1) {
    if (tid < off) { rs[tid] += rs[tid + off]; rq[tid] += rq[tid + off]; }
    __syncthreads();
  }
  if (tid == 0) {
    float m   = rs[0] / (float)nrows;
    float var = rq[0] / (float)nrows - m * m;
    float rst = rsqrtf(var + EPSV);
    float sc  = g[c] * rst;
    scale[c] = sc;
    shift[c] = b[c] - m * sc;
  }
}

__global__ void bn_silu_inplace(float* __restrict__ x, const float* __restrict__ scale,
                                const float* __restrict__ shift, int C, int total) {
  int i = blockIdx.x * blockDim.x + threadIdx.x;
  if (i >= total) return;
  int c = i % C;
  x[i] = silu_(x[i] * scale[c] + shift[c]);
}

// depthwise conv K=5, 'same' pad=2; input is silu(bn(h0)) computed on the fly
__global__ void dwconv_front(const float* __restrict__ h0, const float* __restrict__ scale,
                             const float* __restrict__ shift, const float* __restrict__ w,
                             float* __restrict__ out) {
  int i = blockIdx.x * blockDim.x + threadIdx.x;
  if (i >= NROWS * HID) return;
  const int ch = i % HID;
  const int row = i / HID;
  const int l = row % LL;
  const int base = row - l;
  const float sc = scale[ch], sh = shift[ch];
  float acc = 0.0f;
#pragma unroll
  for (int k = 0; k < KK; ++k) {
    int ll = l + k - 2;
    if (ll >= 0 && ll < LL) {
      float h = h0[(size_t)(base + ll) * HID + ch];
      acc += w[ch * KK + k] * silu_(h * sc + sh);
    }
  }
  out[i] = acc;
}

// depthwise causal conv K=4, pad (3,0), + bias, + silu
__global__ void dwconv_causal_silu(const float* __restrict__ xm, const float* __restrict__ w,
                                   const float* __restrict__ bias, float* __restrict__ out) {
  int i = blockIdx.x * blockDim.x + threadIdx.x;
  if (i >= NROWS * DI) return;
  const int d = i % DI;
  const int row = i / DI;
  const int l = row % LL;
  const int base = row - l;
  float acc = bias[d];
#pragma unroll
  for (int k = 0; k < DCONV; ++k) {
    int ll = l - 3 + k;
    if (ll >= 0) acc += w[d * DCONV + k] * xm[(size_t)(base + ll) * DI + d];
  }
  out[i] = silu_(acc);
}

// x1 = x + bn(e); row LN stats; xn = ln(x1)*mg+mb.  grid = NROWS, block = 256.
__global__ void rows_x1_xn(const float* __restrict__ x, const float* __restrict__ e,
                           const float* __restrict__ scale, const float* __restrict__ shift,
                           const float* __restrict__ mg, const float* __restrict__ mb,
                           float* __restrict__ x1, float* __restrict__ xn,
                           float* __restrict__ rowstats) {
  const int row = blockIdx.x, tid = threadIdx.x;
  const size_t off = (size_t)row * DD;
  float vals[3], s = 0.0f, q = 0.0f;
#pragma unroll
  for (int j = 0; j < 3; ++j) {
    int c = tid + j * 256;
    float v = x[off + c] + e[off + c] * scale[c] + shift[c];
    vals[j] = v; x1[off + c] = v; s += v; q += v * v;
  }
  __shared__ float rs[256], rq[256];
  rs[tid] = s; rq[tid] = q; __syncthreads();
  for (int o = 128; o > 0; o >>= 1) {
    if (tid < o) { rs[tid] += rs[tid + o]; rq[tid] += rq[tid + o]; }
    __syncthreads();
  }
  __shared__ float mS, rS;
  if (tid == 0) {
    float m = rs[0] / (float)DD;
    float var = rq[0] / (float)DD - m * m;
    float rst = rsqrtf(var + EPSV);
    mS = m; rS = rst;
    rowstats[row * 2] = m; rowstats[row * 2 + 1] = rst;
  }
  __syncthreads();
  float m = mS, rst = rS;
#pragma unroll
  for (int j = 0; j < 3; ++j) {
    int c = tid + j * 256;
    xn[off + c] = (vals[j] - m) * rst * mg[c] + mb[c];
  }
}

// xf = x1 + mo*gate + x; row LN stats.
__global__ void final_rows(const float* __restrict__ x1, const float* __restrict__ x,
                           const float* __restrict__ mo, const float* __restrict__ gate,
                           float* __restrict__ xf, float* __restrict__ rowstats) {
  const int row = blockIdx.x, tid = threadIdx.x;
  const size_t off = (size_t)row * DD;
  float s = 0.0f, q = 0.0f;
#pragma unroll
  for (int j = 0; j < 3; ++j) {
    int c = tid + j * 256;
    float v = x1[off + c] + mo[off + c] * gate[off + c] + x[off + c];
    xf[off + c] = v; s += v; q += v * v;
  }
  __shared__ float rs[256], rq[256];
  rs[tid] = s; rq[tid] = q; __syncthreads();
  for (int o = 128; o > 0; o >>= 1) {
    if (tid < o) { rs[tid] += rs[tid + o]; rq[tid] += rq[tid + o]; }
    __syncthreads();
  }
  if (tid == 0) {
    float m = rs[0] / (float)DD;
    float var = rq[0] / (float)DD - m * m;
    rowstats[row * 2] = m;
    rowstats[row * 2 + 1] = rsqrtf(var + EPSV);
  }
}

// feat_attr[b,d] = max_l ( (x1-m)*rstd*g + b )
__global__ void feat_attr_kernel(const float* __restrict__ x1, const float* __restrict__ rs,
                                 const float* __restrict__ g, const float* __restrict__ bb,
                                 float* __restrict__ out) {
  int i = blockIdx.x * blockDim.x + threadIdx.x;
  if (i >= BB * DD) return;
  const int b = i / DD, d = i % DD;
  const float gm = g[d], bm = bb[d];
  float best = -INFINITY;
  for (int l = 0; l < LL; ++l) {
    int row = b * LL + l;
    float v = (x1[(size_t)row * DD + d] - rs[row * 2]) * rs[row * 2 + 1];
    best = fmaxf(best, v * gm + bm);
  }
  out[i] = best;
}

// feat_id[b,d] = mean_l ( (xf-m)*rstd*g + b )
__global__ void feat_id_kernel(const float* __restrict__ xf, const float* __restrict__ rs,
                               const float* __restrict__ g, const float* __restrict__ bb,
                               float* __restrict__ out) {
  int i = blockIdx.x * blockDim.x + threadIdx.x;
  if (i >= BB * DD) return;
  const int b = i / DD, d = i % DD;
  const float gm = g[d], bm = bb[d];
  float acc = 0.0f;
  for (int l = 0; l < LL; ++l) {
    int row = b * LL + l;
    float v = (xf[(size_t)row * DD + d] - rs[row * 2]) * rs[row * 2 + 1];
    acc += v * gm + bm;
  }
  out[i] = acc * (1.0f / (float)LL);
}

// 1D batchnorm over batch axis of feat_id [B, D]. one block, 768 threads.
__global__ void idbn_kernel(const float* __restrict__ fid, const float* __restrict__ g,
                            const float* __restrict__ bb, float* __restrict__ out) {
  int d = threadIdx.x;
  if (d >= DD) return;
  float s = 0.0f, q = 0.0f;
  for (int b = 0; b < BB; ++b) { float v = fid[b * DD + d]; s += v; q += v * v; }
  float m = s / (float)BB;
  float var = q / (float)BB - m * m;
  float rst = rsqrtf(var + EPSV);
  float sc = g[d] * rst, sh = bb[d] - m * sc;
  for (int b = 0; b < BB; ++b) out[b * DD + d] = fid[b * DD + d] * sc + sh;
}

// ---------------------------------------------------------------------------
// Selective scan. one lane per (b, d); 16 SSM states in registers.
// Bm/Cm staged through LDS in 64-timestep chunks; fused x silu(z); y in place.
// ---------------------------------------------------------------------------
__global__ void sel_scan(const float* __restrict__ dbc, const float* __restrict__ delta,
                         const float* __restrict__ Alog, const float* __restrict__ Dp,
                         const float* __restrict__ z, float* __restrict__ uy) {
  const int d = blockIdx.x * blockDim.x + threadIdx.x;
  const int b = blockIdx.y;
  const int tid = threadIdx.x;

  float Ad[DS];
#pragma unroll
  for (int s = 0; s < DS; ++s) Ad[s] = -__expf(Alog[d * DS + s]);
  const float dpar = Dp[d];

  float h[DS];
#pragma unroll
  for (int s = 0; s < DS; ++s) h[s] = 0.0f;

  __shared__ float sBC[64 * 32];

  for (int tc = 0; tc < LL; tc += 64) {
    for (int e = tid; e < 64 * 32; e += 256) {
      int tl = e >> 5, j = e & 31;
      sBC[e] = dbc[(size_t)(b * LL + tc + tl) * 80 + 48 + j];
    }
    __syncthreads();
    for (int tl = 0; tl < 64; ++tl) {
      const size_t idx = (size_t)(b * LL + tc + tl) * DI + d;
      const float dt = delta[idx];
      const float u  = uy[idx];
      const float du = dt * u;
      const float* bc = &sBC[tl * 32];
      float acc = 0.0f;
#pragma unroll
      for (int s = 0; s < DS; ++s) {
        h[s] = h[s] * __expf(dt * Ad[s]) + du * bc[s];
        acc += h[s] * bc[16 + s];
      }
      float y = acc + u * dpar;
      uy[idx] = y * silu_(z[idx]);
    }
    __syncthreads();
  }
}

// ---------------------------------------------------------------------------
// Host side
// ---------------------------------------------------------------------------
static void launch_gemm(const float* A, const float* W, float* C, int M, int N, int K,
                        int lda, int ldw, int ldc, const float* bias, int epi,
                        hipStream_t s) {
  if ((N % BN) == 0 && (K % KC) == 0 && (M % BM) == 0) {
    dim3 g(N / BN, M / BM);
    gemm_f32_wmma_lds<<<g, 256, 0, s>>>(A, W, C, K, lda, ldw, ldc, bias, epi);
  } else if ((N % 64) == 0) {
    dim3 g(N / 64, M / 16);
    gemm_f32_wmma_pipe<4><<<g, 32, 0, s>>>(A, W, C, K, lda, ldw, ldc, bias, epi);
  } else {
    dim3 g(N / 16, M / 16);
    gemm_f32_wmma_pipe<1><<<g, 32, 0, s>>>(A, W, C, K, lda, ldw, ldc, bias, epi);
  }
}

extern "C" void kernel_launch(void* const* d_in, const int* in_sizes, int n_in,
                              void* d_out, int out_size, void* d_ws, size_t ws_size,
                              hipStream_t stream) {
  const float* x        = (const float*)d_in[0];
  const float* sq_w     = (const float*)d_in[1];
  const float* sq_bn_g  = (const float*)d_in[2];
  const float* sq_bn_b  = (const float*)d_in[3];
  const float* dw_w     = (const float*)d_in[4];
  const float* dw_bn_g  = (const float*)d_in[5];
  const float* dw_bn_b  = (const float*)d_in[6];
  const float* ex_w     = (const float*)d_in[7];
  const float* ex_bn_g  = (const float*)d_in[8];
  const float* ex_bn_b  = (const float*)d_in[9];
  const float* attr_g   = (const float*)d_in[10];
  const float* attr_b   = (const float*)d_in[11];
  const float* mnorm_g  = (const float*)d_in[12];
  const float* mnorm_b  = (const float*)d_in[13];
  const float* in_proj_w= (const float*)d_in[14];
  const float* conv_w   = (const float*)d_in[15];
  const float* conv_b   = (const float*)d_in[16];
  const float* xproj_w  = (const float*)d_in[17];
  const float* dtproj_w = (const float*)d_in[18];
  const float* dtproj_b = (const float*)d_in[19];
  const float* A_log    = (const float*)d_in[20];
  const float* D_param  = (const float*)d_in[21];
  const float* out_proj_w = (const float*)d_in[22];
  const float* gate_w   = (const float*)d_in[23];
  const float* gate_b   = (const float*)d_in[24];
  const float* idn_g    = (const float*)d_in[25];
  const float* idn_b    = (const float*)d_in[26];
  const float* idbn_g   = (const float*)d_in[27];
  const float* idbn_b   = (const float*)d_in[28];

  float* out = (float*)d_out;
  float* ws  = (float*)d_ws;

  const size_t szND  = (size_t)NROWS * DD;
  const size_t szNDI = (size_t)NROWS * DI;
  float* x1    = ws;                       // [N, D]
  float* xn    = x1 + szND;                // [N, D] (reused as x_final)
  float* Zbuf  = xn + szND;                // [N, DI]
  float* Abuf  = Zbuf + szNDI;             // [N, DI] xm_raw -> delta (h0 alias)
  float* Ubuf  = Abuf + szNDI;             // [N, DI] xm_conv -> y    (c alias)
  float* dbc   = Ubuf + szNDI;             // [N, 80]
  float* mo    = dbc + (size_t)NROWS * 80; // [N, D] mamba_out        (e alias)
  float* gate  = mo + szND;                // [N, D]
  float* rs1   = gate + szND;              // [N, 2]
  float* rs2   = rs1 + 2 * NROWS;          // [N, 2]
  float* stats = rs2 + 2 * NROWS;
  float* scale0 = stats,        *shift0 = stats + 192;
  float* scale1 = stats + 384,  *shift1 = stats + 576;
  float* scale2 = stats + 1024, *shift2 = stats + 1792;

  float* h0 = Abuf;   // [N, HID]
  float* cb = Ubuf;   // [N, HID]
  float* eb = mo;     // [N, D]

  const int BD = BB * DD;

  // ---- squeeze-excite conv front-end -------------------------------------
  launch_gemm(x, sq_w, h0, NROWS, HID, DD, DD, DD, HID, nullptr, 0, stream);
  bn_stats<<<HID, 256, 0, stream>>>(h0, HID, NROWS, sq_bn_g, sq_bn_b, scale0, shift0);
  {
    int total = NROWS * HID;
    dwconv_front<<<(total + 255) / 256, 256, 0, stream>>>(h0, scale0, shift0, dw_w, cb);
    bn_stats<<<HID, 256, 0, stream>>>(cb, HID, NROWS, dw_bn_g, dw_bn_b, scale1, shift1);
    bn_silu_inplace<<<(total + 255) / 256, 256, 0, stream>>>(cb, scale1, shift1, HID, total);
  }
  launch_gemm(cb, ex_w, eb, NROWS, DD, HID, HID, HID, DD, nullptr, 0, stream);
  bn_stats<<<DD, 256, 0, stream>>>(eb, DD, NROWS, ex_bn_g, ex_bn_b, scale2, shift2);

  // ---- x1 / layernorms / feat_attr ---------------------------------------
  rows_x1_xn<<<NROWS, 256, 0, stream>>>(x, eb, scale2, shift2, mnorm_g, mnorm_b,
                                        x1, xn, rs1);
  feat_attr_kernel<<<(BD + 255) / 256, 256, 0, stream>>>(x1, rs1, attr_g, attr_b, out);

  // ---- Mamba block --------------------------------------------------------
  launch_gemm(xn, in_proj_w,                   Abuf, NROWS, DI, DD, DD, DD, DI, nullptr, 0, stream);
  launch_gemm(xn, in_proj_w + (size_t)DI * DD, Zbuf, NROWS, DI, DD, DD, DD, DI, nullptr, 0, stream);
  {
    int total = NROWS * DI;
    dwconv_causal_silu<<<(total + 255) / 256, 256, 0, stream>>>(Abuf, conv_w, conv_b, Ubuf);
  }
  launch_gemm(Ubuf, xproj_w, dbc, NROWS, 80, DI, DI, DI, 80, nullptr, 0, stream);
  launch_gemm(dbc, dtproj_w, Abuf, NROWS, DI, DTR, 80, DTR, DI, dtproj_b, 1, stream);
  sel_scan<<<dim3(DI / 256, BB), 256, 0, stream>>>(dbc, Abuf, A_log, D_param, Zbuf, Ubuf);
  launch_gemm(Ubuf, out_proj_w, mo, NROWS, DD, DI, DI, DI, DD, nullptr, 0, stream);
  launch_gemm(xn, gate_w, gate, NROWS, DD, DD, DD, DD, DD, gate_b, 2, stream);

  // ---- final residual + pooled outputs ------------------------------------
  final_rows<<<NROWS, 256, 0, stream>>>(x1, x, mo, gate, xn /*=x_final*/, rs2);
  feat_id_kernel<<<(BD + 255) / 256, 256, 0, stream>>>(xn, rs2, idn_g, idn_b, out + BD);
  idbn_kernel<<<1, DD, 0, stream>>>(out + BD, idbn_g, idbn_b, out + 2 * BD);

  (void)in_sizes; (void)n_in; (void)out_size; (void)ws_size;
}